// VisionStateSpaceModule_28089086116478
// MI455X (gfx1250) — compile-verified
//
#include <hip/hip_runtime.h>
#include <hip/hip_bf16.h>
#include <math.h>

// ---------------------------------------------------------------------------
// VisionStateSpaceModule for MI455X (gfx1250), wave32.
// fp32 everywhere; all dense GEMMs use V_WMMA_F32_16X16X4_F32 (fp32 tensor op)
// so the matrix path runs on the WMMA pipe without precision loss.
// Each wave computes a 16x32 output strip: one A fragment feeds two WMMAs.
// ---------------------------------------------------------------------------

typedef __attribute__((ext_vector_type(2))) float v2f;
typedef __attribute__((ext_vector_type(8))) float v8f;

#define Bn 8
#define Hh 64
#define Ww 64
#define Cc 128
#define Dd 128
#define Ll 4096          // H*W
#define BL 32768         // B*L
#define Nst 8
#define Rr 8

__device__ __forceinline__ float silu_f(float v) {
    return v / (1.f + __expf(-v));
}

__device__ __forceinline__ v8f wmma4(v2f a, v2f b, v8f c) {
    return __builtin_amdgcn_wmma_f32_16x16x4_f32(
        false, a, false, b, (short)0, c, false, false);
}

// ---------------------------------------------------------------------------
// C = A(M,K) * B(N,K)^T, one wave per 16x32 strip (two 16x16 tiles sharing
// the A fragment). K compile-time -> fully unrolled (K/4 * 2 WMMAs/wave).
// Fragment layout (ISA 7.12.2, 32-bit A 16x4): lane[3:0] = M row,
// lane[4] selects K pair {0,1} vs {2,3}; B mirrors with lane[3:0] = N col.
// Grids are exact: EXEC all-1s at every WMMA.
// ---------------------------------------------------------------------------
template <int K, int NTOT>
__global__ void k_gemm_nt(const float* __restrict__ A,
                          const float* __restrict__ Bw,
                          float* __restrict__ C) {
    constexpr int TN2 = NTOT / 32;
    int wave = threadIdx.x >> 5;
    int lane = threadIdx.x & 31;
    int tile = blockIdx.x * 8 + wave;
    int tn = tile % TN2;                // 32-wide column block
    int tm = tile / TN2;
    int m  = lane & 15;
    int kh = (lane >> 4) * 2;

    const float* Arow  = A  + (size_t)(tm * 16 + m) * K;
    const float* Brow0 = Bw + (size_t)(tn * 32 + m) * K;
    const float* Brow1 = Brow0 + (size_t)16 * K;

    v8f acc0 = {}, acc1 = {};
#pragma unroll
    for (int k0 = 0; k0 < K; k0 += 4) {
        v2f a  = *(const v2f*)(Arow  + k0 + kh);
        v2f b0 = *(const v2f*)(Brow0 + k0 + kh);
        v2f b1 = *(const v2f*)(Brow1 + k0 + kh);
        acc0 = wmma4(a, b0, acc0);
        acc1 = wmma4(a, b1, acc1);
    }
    int rbase = tm * 16 + ((lane >> 4) * 8);
    int col   = tn * 32 + (lane & 15);
#pragma unroll
    for (int r = 0; r < 8; ++r) {
        C[(size_t)(rbase + r) * NTOT + col]      = acc0[r];
        C[(size_t)(rbase + r) * NTOT + col + 16] = acc1[r];
    }
}

// ---------------------------------------------------------------------------
// Grouped 1x1 conv: 4 independent (M,32)x(32,32)^T GEMMs. One wave covers a
// full group row strip (16x32), 16 WMMAs fully unrolled. Optional fused SiLU.
// tiles = {tm[0,2048), g[0,4)} -> 8192 tiles, 1024 blocks exact.
// ---------------------------------------------------------------------------
template <int SILU>
__global__ void k_grouped_gemm(const float* __restrict__ X,
                               const float* __restrict__ Wg,
                               float* __restrict__ Out) {
    int wave = threadIdx.x >> 5;
    int lane = threadIdx.x & 31;
    int tile = blockIdx.x * 8 + wave;
    int g  = tile & 3;
    int tm = tile >> 2;
    int m  = lane & 15;
    int kh = (lane >> 4) * 2;

    const float* Arow  = X  + (size_t)(tm * 16 + m) * 128 + g * 32;
    const float* Brow0 = Wg + g * 1024 + (size_t)m * 32;
    const float* Brow1 = Brow0 + 16 * 32;

    v8f acc0 = {}, acc1 = {};
#pragma unroll
    for (int k0 = 0; k0 < 32; k0 += 4) {
        v2f a  = *(const v2f*)(Arow  + k0 + kh);
        v2f b0 = *(const v2f*)(Brow0 + k0 + kh);
        v2f b1 = *(const v2f*)(Brow1 + k0 + kh);
        acc0 = wmma4(a, b0, acc0);
        acc1 = wmma4(a, b1, acc1);
    }
    int rbase = tm * 16 + ((lane >> 4) * 8);
    int col   = g * 32 + (lane & 15);
#pragma unroll
    for (int r = 0; r < 8; ++r) {
        float v0 = acc0[r], v1 = acc1[r];
        if (SILU) { v0 = silu_f(v0); v1 = silu_f(v1); }
        Out[(size_t)(rbase + r) * 128 + col]      = v0;
        Out[(size_t)(rbase + r) * 128 + col + 16] = v1;
    }
}

// ---------------------------------------------------------------------------
// Depthwise 3x3 SAME conv + SiLU. One thread per (b,h,w,d).
// ---------------------------------------------------------------------------
__global__ void k_dwconv_silu(const float* __restrict__ X,
                              const float* __restrict__ Wd,
                              float* __restrict__ Out) {
    int idx = blockIdx.x * 256 + threadIdx.x;   // B*H*W*128 = 4194304
    int d = idx & 127;
    int w = (idx >> 7) & 63;
    int h = (idx >> 13) & 63;
    int b = idx >> 19;
    float acc = 0.f;
#pragma unroll
    for (int ky = 0; ky < 3; ++ky) {
        int hh = h + ky - 1;
        if ((unsigned)hh >= 64u) continue;
#pragma unroll
        for (int kx = 0; kx < 3; ++kx) {
            int ww = w + kx - 1;
            if ((unsigned)ww >= 64u) continue;
            acc += X[(((size_t)b * 64 + hh) * 64 + ww) * 128 + d] *
                   Wd[(ky * 3 + kx) * 128 + d];
        }
    }
    Out[idx] = silu_f(acc);
}

// ---------------------------------------------------------------------------
// Causal depthwise conv1d (width 3) over L on the u half of xz, + bias + SiLU.
// ---------------------------------------------------------------------------
__global__ void k_conv1d_silu(const float* __restrict__ XZ,
                              const float* __restrict__ Wc,
                              const float* __restrict__ Bc,
                              float* __restrict__ U) {
    int idx = blockIdx.x * 256 + threadIdx.x;   // B*L*128
    int d = idx & 127;
    int l = (idx >> 7) & 4095;
    int b = idx >> 19;
    size_t rb = (size_t)b * Ll;
    float acc = Bc[d];
    if (l >= 2) acc += XZ[(rb + l - 2) * 256 + d] * Wc[d * 3 + 0];
    if (l >= 1) acc += XZ[(rb + l - 1) * 256 + d] * Wc[d * 3 + 1];
    acc += XZ[(rb + l) * 256 + d] * Wc[d * 3 + 2];
    U[idx] = silu_f(acc);
}

// ---------------------------------------------------------------------------
// x_proj: (BL,128) x (24,128)^T. Tiny N=24, plain VALU dot products.
// ---------------------------------------------------------------------------
__global__ void k_xproj(const float* __restrict__ U,
                        const float* __restrict__ Wp,
                        float* __restrict__ Xd) {
    int idx = blockIdx.x * 256 + threadIdx.x;   // BL*24 = 786432
    int e   = idx % 24;
    int row = idx / 24;
    const float* ur = U  + (size_t)row * 128;
    const float* wr = Wp + e * 128;
    float acc = 0.f;
#pragma unroll 8
    for (int k = 0; k < 128; ++k) acc += ur[k] * wr[k];
    Xd[idx] = acc;
}

// ---------------------------------------------------------------------------
// dt_proj + softplus -> delta (BL,128)
// ---------------------------------------------------------------------------
__global__ void k_dtproj(const float* __restrict__ Xd,
                         const float* __restrict__ Wdt,
                         const float* __restrict__ Bdt,
                         float* __restrict__ Delta) {
    int idx = blockIdx.x * 256 + threadIdx.x;   // B*L*128
    int d   = idx & 127;
    int row = idx >> 7;
    const float* dt = Xd + (size_t)row * 24;
    float acc = Bdt[d];
#pragma unroll
    for (int r = 0; r < Rr; ++r) acc += dt[r] * Wdt[d * Rr + r];
    Delta[idx] = (acc > 20.f) ? acc : log1pf(__expf(acc));
}

// ---------------------------------------------------------------------------
// Selective scan. One lane per (b,d,n): 8192 independent recurrences =
// 256 waves, each doing 4096 dependent exp->fma steps (the latency-critical
// path of the whole module). N=8 reduced with three shfl_xor steps; lane n==0
// applies the D-residual + z-gate and writes y. global_prefetch_b8 keeps the
// delta/u/B/C streams 16 rows ahead in the near cache so the only loop-carried
// latency is v_exp -> v_fma. Block 256 = 32 d x 8 n; grid 32 = B * 4 d-blocks.
// ---------------------------------------------------------------------------
__global__ void k_scan(const float* __restrict__ Delta,
                       const float* __restrict__ U,
                       const float* __restrict__ Xd,
                       const float* __restrict__ XZ,
                       const float* __restrict__ Alog,
                       const float* __restrict__ Dp,
                       float* __restrict__ Y) {
    int tid = threadIdx.x;
    int n   = tid & 7;
    int dl  = tid >> 3;                    // 0..31
    int b   = blockIdx.x >> 2;
    int d   = ((blockIdx.x & 3) << 5) + dl;
    float A  = -__expf(Alog[d * Nst + n]);
    float Dv = Dp[d];
    float h  = 0.f;
    size_t rb = (size_t)b * Ll;
    for (int l = 0; l < Ll; ++l) {
        size_t row = rb + l;
        if ((l & 3) == 0 && l + 16 < Ll) {       // uniform branch
            size_t prow = row + 16;
            __builtin_prefetch(Delta + prow * 128 + d, 0, 3);
            __builtin_prefetch(U     + prow * 128 + d, 0, 3);
            __builtin_prefetch(Xd    + prow * 24,      0, 3);
            __builtin_prefetch(XZ    + prow * 256 + 128 + d, 0, 3);
        }
        float dv = Delta[row * 128 + d];
        float uv = U[row * 128 + d];
        float Bv = Xd[row * 24 + Rr + n];
        float Cv = Xd[row * 24 + Rr + Nst + n];
        h = __expf(dv * A) * h + (dv * uv) * Bv;
        float y = h * Cv;
        y += __shfl_xor(y, 1, 32);
        y += __shfl_xor(y, 2, 32);
        y += __shfl_xor(y, 4, 32);
        if (n == 0) {
            float z = XZ[row * 256 + 128 + d];
            Y[row * 128 + d] = (y + uv * Dv) * silu_f(z);
        }
    }
}

// ---------------------------------------------------------------------------
// LayerNorm over D=128 (one wave per row, 4 channels/lane, shfl reduction),
// fused with the x2 gate multiply.
// ---------------------------------------------------------------------------
__global__ void k_ln_gate(const float* __restrict__ Xm,
                          const float* __restrict__ X2,
                          const float* __restrict__ Gamma,
                          const float* __restrict__ Beta,
                          float* __restrict__ Out) {
    int wave = threadIdx.x >> 5;
    int lane = threadIdx.x & 31;
    int row  = blockIdx.x * 8 + wave;      // 4096 blocks * 8 = BL rows
    const float* xr = Xm + (size_t)row * 128;
    float v[4];
    float s = 0.f, s2 = 0.f;
#pragma unroll
    for (int i = 0; i < 4; ++i) {
        v[i] = xr[lane * 4 + i];
        s += v[i];
        s2 += v[i] * v[i];
    }
#pragma unroll
    for (int off = 16; off > 0; off >>= 1) {
        s  += __shfl_xor(s,  off, 32);
        s2 += __shfl_xor(s2, off, 32);
    }
    float mu  = s * (1.f / 128.f);
    float var = s2 * (1.f / 128.f) - mu * mu;
    float rs  = rsqrtf(var + 1e-5f);
#pragma unroll
    for (int i = 0; i < 4; ++i) {
        int c = lane * 4 + i;
        float o = (v[i] - mu) * rs * Gamma[c] + Beta[c];
        Out[(size_t)row * 128 + c] = o * X2[(size_t)row * 128 + c];
    }
}

// ---------------------------------------------------------------------------
// Launch. Workspace carve (floats), ~104 MB with reuse (fits in the 192 MB
// global L2, so the inter-kernel traffic never leaves the chip):
//   xz   : [0,        8388608)   (BL x 256)  live k4..k8
//   t1   : [8388608,  12582912)  (BL x 128)  t1 -> delta -> x_fuse
//   x2   : [12582912, 16777216)  live k2..k10
//   x1s  : [16777216, 20971520)  x1s -> y
//   ucv  : [20971520, 25165824)  u_conv -> out_proj result
//   xdbl : [25165824, 25952256)  (BL x 24)
// ---------------------------------------------------------------------------
extern "C" void kernel_launch(void* const* d_in, const int* in_sizes, int n_in,
                              void* d_out, int out_size, void* d_ws, size_t ws_size,
                              hipStream_t stream) {
    const float* x         = (const float*)d_in[0];
    const float* w1        = (const float*)d_in[1];
    const float* dw        = (const float*)d_in[2];
    const float* in_proj_w = (const float*)d_in[3];
    const float* conv1d_w  = (const float*)d_in[4];
    const float* conv1d_b  = (const float*)d_in[5];
    const float* x_proj_w  = (const float*)d_in[6];
    const float* dt_proj_w = (const float*)d_in[7];
    const float* dt_proj_b = (const float*)d_in[8];
    const float* A_log     = (const float*)d_in[9];
    const float* D_param   = (const float*)d_in[10];
    const float* out_proj_w= (const float*)d_in[11];
    const float* gamma     = (const float*)d_in[12];
    const float* beta      = (const float*)d_in[13];
    const float* w2        = (const float*)d_in[14];
    const float* wout      = (const float*)d_in[15];
    float* out = (float*)d_out;

    float* ws   = (float*)d_ws;
    float* xz   = ws;                       // BL*256
    float* t1   = ws + (size_t)BL * 256;    // BL*128
    float* x2   = t1 + (size_t)BL * 128;    // BL*128
    float* x1s  = x2 + (size_t)BL * 128;    // BL*128
    float* ucv  = x1s + (size_t)BL * 128;   // BL*128
    float* xdbl = ucv + (size_t)BL * 128;   // BL*24
    float* delta = t1;                      // reuse
    float* y     = x1s;                     // reuse
    float* x1m   = ucv;                     // reuse
    float* xfuse = t1;                      // reuse

    // 1. x1 = grouped_1x1(x, w1)
    k_grouped_gemm<0><<<1024, 256, 0, stream>>>(x, w1, t1);
    // 2. x2 = silu(grouped_1x1(x, w2))
    k_grouped_gemm<1><<<1024, 256, 0, stream>>>(x, w2, x2);
    // 3. x1 = silu(dwconv3x3(x1))
    k_dwconv_silu<<<16384, 256, 0, stream>>>(t1, dw, x1s);
    // 4. xz = x1_seq @ in_proj_w^T   (WMMA, M=32768 N=256 K=128)
    k_gemm_nt<128, 256><<<2048, 256, 0, stream>>>(x1s, in_proj_w, xz);
    // 5. u = silu(causal_conv1d(u) + b)
    k_conv1d_silu<<<16384, 256, 0, stream>>>(xz, conv1d_w, conv1d_b, ucv);
    // 6. x_dbl = u @ x_proj_w^T
    k_xproj<<<3072, 256, 0, stream>>>(ucv, x_proj_w, xdbl);
    // 7. delta = softplus(dt @ dt_proj_w^T + b)
    k_dtproj<<<16384, 256, 0, stream>>>(xdbl, dt_proj_w, dt_proj_b, delta);
    // 8. selective scan + D residual + z gate
    k_scan<<<32, 256, 0, stream>>>(delta, ucv, xdbl, xz, A_log, D_param, y);
    // 9. x1m = y @ out_proj_w^T   (WMMA, M=32768 N=128 K=128)
    k_gemm_nt<128, 128><<<1024, 256, 0, stream>>>(y, out_proj_w, x1m);
    // 10. layernorm * gamma + beta, * x2
    k_ln_gate<<<4096, 256, 0, stream>>>(x1m, x2, gamma, beta, xfuse);
    // 11. out = grouped_1x1(x_fuse, wout)
    k_grouped_gemm<0><<<1024, 256, 0, stream>>>(xfuse, wout, out);
}